// Net_64484638982411
// MI455X (gfx1250) — compile-verified
//
#include <hip/hip_runtime.h>

// ---------------------------------------------------------------------------
// GNN: h = relu(x@W1+b1); agg = segment_mean(h[src] -> dst);
//      h2 = relu(agg@Wl + bl + h@Wr); out = h2@W2 + b2
// f32 everywhere -> V_WMMA_F32_16X16X4_F32 (exact f32 matrix op, wave32).
// Store guards are tile-uniform (wave-uniform branch) so epilogues are
// straight-line store clauses with full EXEC.
// ---------------------------------------------------------------------------

typedef float v2f __attribute__((ext_vector_type(2)));
typedef float v8f __attribute__((ext_vector_type(8)));

__device__ __forceinline__ v8f wmma_f32_16x16x4(v2f a, v2f b, v8f c) {
    // (neg_a, A, neg_b, B, c_mod, C, reuse_a, reuse_b)
    return __builtin_amdgcn_wmma_f32_16x16x4_f32(false, a, false, b,
                                                 (short)0, c, false, false);
}

// B fragment: 4x16 slice of a KxNcols row-major weight.
// VGPR n holds row K=k0+n (lanes 0-15) / K=k0+n+2 (lanes 16-31), col = lane&15.
__device__ __forceinline__ v2f load_bfrag(const float* __restrict__ w, int ncols,
                                          int k0, int hiK, int col) {
    v2f b;
    b.x = w[(k0 + hiK + 0) * ncols + col];
    b.y = w[(k0 + hiK + 1) * ncols + col];
    return b;
}

// ---------------- Kernel 1: h = relu(x @ lin1_w + lin1_b) -------------------
__global__ __launch_bounds__(256) void lin1_relu_kernel(
    const float* __restrict__ x, const float* __restrict__ w,
    const float* __restrict__ b, float* __restrict__ h, int n, int nTiles) {
    const int lane = threadIdx.x & 31;
    const int wave = threadIdx.x >> 5;
    const int tile = blockIdx.x * 8 + wave;
    if (tile >= nTiles) return;            // wave-uniform: EXEC stays all-1s

    const int col  = lane & 15;
    const int hiK  = (lane >> 4) * 2;      // 0 for lanes 0-15, 2 for 16-31
    int row  = (tile << 4) + col;          // A-matrix row for this lane
    int rowL = row < n ? row : n - 1;      // clamp loads (keep EXEC full)

    v8f acc = {};
#pragma unroll
    for (int k0 = 0; k0 < 16; k0 += 4) {
        v2f a = *(const v2f*)(x + (size_t)rowL * 16 + k0 + hiK);
        acc = wmma_f32_16x16x4(a, load_bfrag(w, 16, k0, hiK, col), acc);
    }
    const float bias = b[col];
    const int mBase  = (tile << 4) + ((lane >> 4) << 3);  // +8 rows hi half
    float* hp = h + (size_t)mBase * 16 + col;
    if ((tile << 4) + 16 <= n) {           // wave-uniform full-tile fast path
#pragma unroll
        for (int i = 0; i < 8; ++i) {
            float v = acc[i] + bias;
            hp[i * 16] = v > 0.f ? v : 0.f;
        }
    } else {
#pragma unroll
        for (int i = 0; i < 8; ++i) {
            if (mBase + i < n) {
                float v = acc[i] + bias;
                hp[i * 16] = v > 0.f ? v : 0.f;
            }
        }
    }
}

// ---------------- Kernel 2a: zero agg + cnt --------------------------------
__global__ void zero_kernel(float* __restrict__ agg, float* __restrict__ cnt,
                            int n16, int n) {
    int i = blockIdx.x * blockDim.x + threadIdx.x;
    if (i < n16) agg[i] = 0.f;
    if (i < n)   cnt[i] = 0.f;
}

// ---------------- Kernel 2b: edge scatter (mean numerator + counts) --------
// 1 thread per edge: 4x b128 gathers of h[src], 16 f32 atomics + 1 count.
__global__ __launch_bounds__(256) void scatter_kernel(
    const long long* __restrict__ ei, const float* __restrict__ h,
    float* __restrict__ agg, float* __restrict__ cnt, int nEdges) {
    int e = blockIdx.x * blockDim.x + threadIdx.x;
    if (e >= nEdges) return;
    int src = (int)ei[e];
    int dst = (int)ei[(size_t)nEdges + e];
    const float4* hs = (const float4*)(h + (size_t)src * 16);
    float* base = agg + (size_t)dst * 16;
#pragma unroll
    for (int q = 0; q < 4; ++q) {
        float4 v = hs[q];
        atomicAdd(base + q * 4 + 0, v.x);
        atomicAdd(base + q * 4 + 1, v.y);
        atomicAdd(base + q * 4 + 2, v.z);
        atomicAdd(base + q * 4 + 3, v.w);
    }
    atomicAdd(cnt + dst, 1.0f);
}

// ---------------- Kernel 3: SAGE + lin2 (fused) -----------------------------
__global__ __launch_bounds__(256) void sage_lin2_kernel(
    const float* __restrict__ h, const float* __restrict__ agg,
    const float* __restrict__ cnt, const float* __restrict__ wl,
    const float* __restrict__ bl, const float* __restrict__ wr,
    const float* __restrict__ w2, const float* __restrict__ b2,
    float* __restrict__ out, int n, int nTiles) {
    __shared__ float lds[8][16][17];       // per-wave 16x16 tile, padded
    const int lane = threadIdx.x & 31;
    const int wave = threadIdx.x >> 5;
    const int tile = blockIdx.x * 8 + wave;
    if (tile >= nTiles) return;            // wave-uniform

    const int col = lane & 15;
    const int hiK = (lane >> 4) * 2;
    int row  = (tile << 4) + col;
    int rowL = row < n ? row : n - 1;

    float c   = cnt[rowL];
    float inv = 1.0f / (c > 1.0f ? c : 1.0f);

    // acc = mean(agg) @ Wl  +  h @ Wr
    v8f acc = {};
#pragma unroll
    for (int k0 = 0; k0 < 16; k0 += 4) {
        v2f a = *(const v2f*)(agg + (size_t)rowL * 16 + k0 + hiK);
        a.x *= inv; a.y *= inv;
        acc = wmma_f32_16x16x4(a, load_bfrag(wl, 16, k0, hiK, col), acc);
    }
#pragma unroll
    for (int k0 = 0; k0 < 16; k0 += 4) {
        v2f a = *(const v2f*)(h + (size_t)rowL * 16 + k0 + hiK);
        acc = wmma_f32_16x16x4(a, load_bfrag(wr, 16, k0, hiK, col), acc);
    }

    // relu(+bias), park h2 tile in LDS to convert C/D layout -> A layout.
    // Intra-wave only: compiler inserts s_wait_dscnt; no barrier needed and
    // EXEC stays full for the WMMAs below.
    const float biasL = bl[col];
    const int mOff = (lane >> 4) << 3;
#pragma unroll
    for (int i = 0; i < 8; ++i) {
        float v = acc[i] + biasL;
        lds[wave][mOff + i][col] = v > 0.f ? v : 0.f;
    }

    // out = h2 @ W2 + b2, W2 is 16x32 -> two 16x16 WMMA chains
    v8f o0 = {}, o1 = {};
#pragma unroll
    for (int k0 = 0; k0 < 16; k0 += 4) {
        v2f a;
        a.x = lds[wave][col][k0 + hiK + 0];
        a.y = lds[wave][col][k0 + hiK + 1];
        o0 = wmma_f32_16x16x4(a, load_bfrag(w2, 32, k0, hiK, col),      o0);
        o1 = wmma_f32_16x16x4(a, load_bfrag(w2, 32, k0, hiK, col + 16), o1);
    }
    const float bias0 = b2[col];
    const float bias1 = b2[16 + col];
    const int mBase = (tile << 4) + mOff;
    float* op = out + (size_t)mBase * 32 + col;
    if ((tile << 4) + 16 <= n) {           // wave-uniform full-tile fast path
#pragma unroll
        for (int i = 0; i < 8; ++i) {
            op[i * 32]      = o0[i] + bias0;
            op[i * 32 + 16] = o1[i] + bias1;
        }
    } else {
#pragma unroll
        for (int i = 0; i < 8; ++i) {
            if (mBase + i < n) {
                op[i * 32]      = o0[i] + bias0;
                op[i * 32 + 16] = o1[i] + bias1;
            }
        }
    }
}

// ---------------------------------------------------------------------------
extern "C" void kernel_launch(void* const* d_in, const int* in_sizes, int n_in,
                              void* d_out, int out_size, void* d_ws, size_t ws_size,
                              hipStream_t stream) {
    const float*     x      = (const float*)d_in[0];
    const long long* ei     = (const long long*)d_in[1];   // int64 [2,E]
    const float*     lin1_w = (const float*)d_in[2];
    const float*     lin1_b = (const float*)d_in[3];
    const float*     sageWl = (const float*)d_in[4];
    const float*     sageBl = (const float*)d_in[5];
    const float*     sageWr = (const float*)d_in[6];
    const float*     lin2_w = (const float*)d_in[7];
    const float*     lin2_b = (const float*)d_in[8];
    float*           out    = (float*)d_out;

    const int N = in_sizes[0] / 16;   // 100000
    const int E = in_sizes[1] / 2;    // 3200000

    // workspace layout: h[N*16] | agg[N*16] | cnt[N]
    float* h   = (float*)d_ws;
    float* agg = h + (size_t)N * 16;
    float* cnt = agg + (size_t)N * 16;

    const int nTiles   = (N + 15) / 16;
    const int nodeBlks = (nTiles + 7) / 8;          // 8 waves (tiles) per block

    lin1_relu_kernel<<<nodeBlks, 256, 0, stream>>>(x, lin1_w, lin1_b, h, N, nTiles);

    const int n16 = N * 16;
    zero_kernel<<<(n16 + 255) / 256, 256, 0, stream>>>(agg, cnt, n16, N);

    scatter_kernel<<<(E + 255) / 256, 256, 0, stream>>>(ei, h, agg, cnt, E);

    sage_lin2_kernel<<<nodeBlks, 256, 0, stream>>>(h, agg, cnt, sageWl, sageBl,
                                                   sageWr, lin2_w, lin2_b, out,
                                                   N, nTiles);
}